// DilatedGraphConvolutionCell_20435454395079
// MI455X (gfx1250) — compile-verified
//
#include <hip/hip_runtime.h>
#include <hip/hip_bf16.h>

typedef __attribute__((ext_vector_type(2))) float v2f;
typedef __attribute__((ext_vector_type(8))) float v8f;

// D = A(16x4,f32) * B(4x16,f32) + C  -> V_WMMA_F32_16X16X4_F32
__device__ __forceinline__ v8f wmma_f32(v2f a, v2f b, v8f c) {
  return __builtin_amdgcn_wmma_f32_16x16x4_f32(false, a, false, b, (short)0, c,
                                               false, false);
}

// ---- pack obs (1024,8,12) -> OBS (16 x 8192), rows 12..15 zero -----------
__global__ void k_pack_obs(const float* __restrict__ obs, float* __restrict__ O) {
  int idx = blockIdx.x * 256 + threadIdx.x;
  if (idx >= 16 * 8192) return;
  int l = idx >> 13;
  int col = idx & 8191;               // n*8 + d
  float v = 0.f;
  if (l < 12) v = obs[col * 12 + l];
  O[idx] = v;
}

// ---- generic zero-pad pack: src (M x K) -> dst (16 x K) ------------------
__global__ void k_pack_pad(const float* __restrict__ src, float* __restrict__ dst,
                           int M, int K) {
  int idx = blockIdx.x * 256 + threadIdx.x;
  if (idx >= 16 * K) return;
  int r = idx / K;
  dst[idx] = (r < M) ? src[idx] : 0.f;
}

// ---- split-K skinny GEMM: part[s][16][N] = A(16xK chunk) * W(KxN) --------
// All A operands are 16-row buffers; rows >= M carry garbage that only
// pollutes C rows >= M, which are never consumed downstream.
__global__ void __launch_bounds__(32)
k_gemm_wmma(const float* __restrict__ A, int lda,
            const float* __restrict__ W, int N, int K, int klen,
            float* __restrict__ part) {
  int nt = blockIdx.x, s = blockIdx.y;
  int k0 = s * klen;
  int k1 = min(K, k0 + klen);
  int L = threadIdx.x;
  int m = L & 15, h = L >> 4;
  int n = nt * 16 + m;
  const float* Ap = A + m * lda + k0 + 2 * h;
  const float* Wp = W + (size_t)(k0 + 2 * h) * N + n;
  const size_t sW = (size_t)4 * N;    // W rows per k-step
  v8f acc = {};
  int k = k0;
  for (; k + 16 <= k1; k += 16) {
    __builtin_prefetch(Wp + 16 * N, 0, 1);   // 1 iter (16 W rows) ahead on weight stream
    v2f a0, a1, a2, a3, b0, b1, b2, b3;
    a0.x = Ap[0];  a0.y = Ap[1];
    a1.x = Ap[4];  a1.y = Ap[5];
    a2.x = Ap[8];  a2.y = Ap[9];
    a3.x = Ap[12]; a3.y = Ap[13];
    b0.x = Wp[0];           b0.y = Wp[N];
    b1.x = Wp[sW];          b1.y = Wp[sW + N];
    b2.x = Wp[2 * sW];      b2.y = Wp[2 * sW + N];
    b3.x = Wp[3 * sW];      b3.y = Wp[3 * sW + N];
    acc = wmma_f32(a0, b0, acc);
    acc = wmma_f32(a1, b1, acc);
    acc = wmma_f32(a2, b2, acc);
    acc = wmma_f32(a3, b3, acc);
    Ap += 16; Wp += 4 * sW;
  }
  for (; k < k1; k += 4) {
    v2f a, b;
    a.x = Ap[0]; a.y = Ap[1];
    b.x = Wp[0]; b.y = Wp[N];
    acc = wmma_f32(a, b, acc);
    Ap += 4; Wp += sW;
  }
  float* P = part + (size_t)s * 16 * N;
#pragma unroll
  for (int r = 0; r < 8; ++r)
    P[(size_t)(r + 8 * h) * N + n] = acc[r];
}

__global__ void k_reduce_bias_relu(const float* __restrict__ part, int nsplit,
                                   const float* __restrict__ bias, int N,
                                   float* __restrict__ out) {
  int idx = blockIdx.x * 256 + threadIdx.x;
  if (idx >= 16 * N) return;
  int n = idx % N;
  float s = bias[n];
  for (int q = 0; q < nsplit; ++q) s += part[(size_t)q * 16 * N + idx];
  out[idx] = fmaxf(s, 0.f);
}

// ---- build G_a^T (a=9,10,11) and V_10^T from U = SP + TP -----------------
__global__ void k_build_gv(const float* __restrict__ SPU,
                           const float* __restrict__ TPU,
                           const float* __restrict__ Bm,
                           float* __restrict__ GT,    // [3][12][1024]
                           float* __restrict__ VCT) { // [12][1024]
  int idx = blockIdx.x * 256 + threadIdx.x;
  if (idx < 3 * 12 * 1024) {
    int n  = idx & 1023;
    int tp = (idx >> 10) % 12;
    int a  = 9 + idx / (12 * 1024);
    float g = 0.f;
    float tpa = TPU[n * 12 + a];
    for (int t = 0; t < 12; ++t)
      g += (SPU[t * 12288 + n * 12 + a] + tpa) * Bm[t * 12 + tp];
    GT[idx] = g;
  } else if (idx < 3 * 12 * 1024 + 12 * 1024) {
    int r = idx - 3 * 12 * 1024;
    int j = r & 1023, t = r >> 10;
    VCT[r] = SPU[t * 12288 + j * 12 + 10] + TPU[j * 12 + 10];
  }
}

// ---- Y buffers: p0 = X10*(Wf0+Wb0), p1 = X9*Wf1, p2 = X9*Wb1 -------------
__global__ void k_build_y(const float* __restrict__ H3,
                          const float* __restrict__ Wf,
                          const float* __restrict__ Wb,
                          float* __restrict__ Y) { // [3][1024][16]
  int idx = blockIdx.x * 256 + threadIdx.x;
  if (idx >= 3 * 1024 * 16) return;
  int f = idx & 15;
  int n = (idx >> 4) & 1023;
  int p = idx >> 14;
  float v = 0.f;
  if (f < 12) {
    int t = (p == 0) ? 10 : 9;
    const float* X = H3 + t * 12288 + n * 12;
    for (int ff = 0; ff < 12; ++ff) {
      float w;
      if (p == 0)      w = Wf[ff * 12 + f] + Wb[ff * 12 + f];
      else if (p == 1) w = Wf[144 + ff * 12 + f];
      else             w = Wb[144 + ff * 12 + f];
      v += X[ff] * w;
    }
  }
  Y[idx] = v;
}

// ---- fused: P = softmax_rows(thresh(G_a * V_c^T)) * Y, flash-style -------
__global__ void __launch_bounds__(128)
k_adj_softmax_mm(const float* __restrict__ GT,   // [3][12][1024]
                 const float* __restrict__ VCT,  // [12][1024]
                 const float* __restrict__ Y,    // [3][1024][16]
                 float* __restrict__ P) {        // [3][1024][16]
  __shared__ float stage[4][16][17];
  int wid = threadIdx.x >> 5;
  int L = threadIdx.x & 31;
  int gw = blockIdx.x * 4 + wid;        // 0..191 = 3 products x 64 row-blocks
  int prod = gw >> 6;
  int i0 = (gw & 63) * 16;
  int m = L & 15, h = L >> 4;
  int ai = (prod == 0) ? 1 : (prod == 1 ? 0 : 2); // prod->a: 10, 9, 11
  const float* G  = GT + ai * 12 * 1024;
  const float* Yp = Y + prod * 1024 * 16;

  v2f afr[3];
#pragma unroll
  for (int ks = 0; ks < 3; ++ks) {
    int k = ks * 4 + 2 * h;
    afr[ks].x = G[k * 1024 + i0 + m];
    afr[ks].y = G[(k + 1) * 1024 + i0 + m];
  }

  float rowm[8], rsum[8];
#pragma unroll
  for (int r = 0; r < 8; ++r) { rowm[r] = -1e30f; rsum[r] = 0.f; }
  v8f pacc = {};

  for (int j0 = 0; j0 < 1024; j0 += 16) {
    v8f c = {};
#pragma unroll
    for (int ks = 0; ks < 3; ++ks) {
      int k = ks * 4 + 2 * h;
      v2f b;
      b.x = VCT[k * 1024 + j0 + m];
      b.y = VCT[(k + 1) * 1024 + j0 + m];
      c = wmma_f32(afr[ks], b, c);
    }
    float e[8];
#pragma unroll
    for (int r = 0; r < 8; ++r) {
      float x = c[r];
      x = (x >= 0.05f) ? x : 0.f;           // threshold; softmax keeps zeros
      float tmax = x;                        // row stats within 16-lane group
      tmax = fmaxf(tmax, __shfl_xor(tmax, 1, 32));
      tmax = fmaxf(tmax, __shfl_xor(tmax, 2, 32));
      tmax = fmaxf(tmax, __shfl_xor(tmax, 4, 32));
      tmax = fmaxf(tmax, __shfl_xor(tmax, 8, 32));
      float nm = fmaxf(rowm[r], tmax);
      float ex = __expf(x - nm);
      float sum = ex;
      sum += __shfl_xor(sum, 1, 32);
      sum += __shfl_xor(sum, 2, 32);
      sum += __shfl_xor(sum, 4, 32);
      sum += __shfl_xor(sum, 8, 32);
      float scale = __expf(rowm[r] - nm);
      rsum[r] = rsum[r] * scale + sum;
      rowm[r] = nm;
      pacc[r] *= scale;
      e[r] = ex;
    }
    // bounce exp tile through LDS: C-layout -> A-operand layout
#pragma unroll
    for (int r = 0; r < 8; ++r)
      stage[wid][r + 8 * h][m] = e[r];
    asm volatile("s_wait_dscnt 0" ::: "memory");
#pragma unroll
    for (int ks = 0; ks < 4; ++ks) {
      int k = ks * 4 + 2 * h;
      v2f a, b;
      a.x = stage[wid][m][k];
      a.y = stage[wid][m][k + 1];
      b.x = Yp[(j0 + k) * 16 + m];
      b.y = Yp[(j0 + k + 1) * 16 + m];
      pacc = wmma_f32(a, b, pacc);
    }
    asm volatile("s_wait_dscnt 0" ::: "memory"); // stage reuse next iter
  }
#pragma unroll
  for (int r = 0; r < 8; ++r)
    P[prod * 1024 * 16 + (i0 + r + 8 * h) * 16 + m] = pacc[r] / rsum[r];
}

// ---- out0 = relu(P0+b)+relu(P1+P2+b); out1..3 = 0 ------------------------
__global__ void k_epilogue(const float* __restrict__ P,
                           const float* __restrict__ b,
                           float* __restrict__ out) {
  int idx = blockIdx.x * 256 + threadIdx.x;
  if (idx >= 4 * 12288) return;
  float v = 0.f;
  if (idx < 12288) {
    int n = idx / 12, f = idx % 12;
    float p0 = P[n * 16 + f];
    float p1 = P[16384 + n * 16 + f] + P[32768 + n * 16 + f];
    float bb = b[f];
    v = fmaxf(p0 + bb, 0.f) + fmaxf(p1 + bb, 0.f);
  }
  out[idx] = v;
}

extern "C" void kernel_launch(void* const* d_in, const int* in_sizes, int n_in,
                              void* d_out, int out_size, void* d_ws, size_t ws_size,
                              hipStream_t stream) {
  const float* obs   = (const float*)d_in[0];
  const float* tf    = (const float*)d_in[1];
  const float* fc_w1 = (const float*)d_in[2];
  const float* fc_b1 = (const float*)d_in[3];
  const float* fc_w2 = (const float*)d_in[4];
  const float* fc_b2 = (const float*)d_in[5];
  const float* fc_w3 = (const float*)d_in[6];
  const float* fc_b3 = (const float*)d_in[7];
  const float* Wf    = (const float*)d_in[8];
  const float* Wb    = (const float*)d_in[9];
  const float* bvec  = (const float*)d_in[10];
  const float* LI    = (const float*)d_in[11];
  const float* gs_w1 = (const float*)d_in[12];
  const float* gs_b1 = (const float*)d_in[13];
  const float* gs_w2 = (const float*)d_in[14];
  const float* gs_b2 = (const float*)d_in[15];
  const float* gt_w1 = (const float*)d_in[16];
  const float* gt_b1 = (const float*)d_in[17];
  const float* gt_w2 = (const float*)d_in[18];
  const float* gt_b2 = (const float*)d_in[19];
  const float* Bm    = (const float*)d_in[20];

  float* ws = (float*)d_ws;
  size_t o = 0;
  float* OBS = ws + o; o += 16 * 8192;
  float* H1  = ws + o; o += 16 * 2048;
  float* H2  = ws + o; o += 16 * 2048;
  float* H3  = ws + o; o += 16 * 12288;
  float* SP1 = ws + o; o += 16 * 256;
  float* SPU = ws + o; o += 16 * 12288;
  float* TP1 = ws + o; o += 16 * 256;
  float* TPU = ws + o; o += 16 * 12288;
  float* LIP = ws + o; o += 16 * 1024;   // padded layer_initial
  float* TFP = ws + o; o += 16 * 36;     // padded time_features
  float* GT  = ws + o; o += 3 * 12 * 1024;
  float* VCT = ws + o; o += 12 * 1024;
  float* Yb  = ws + o; o += 3 * 1024 * 16;
  float* Pb  = ws + o; o += 3 * 1024 * 16;
  float* part = ws + o; o += (size_t)16 * 16 * 12288 / 4; // 786432 floats

  auto gemm = [&](const float* A, int lda, const float* W,
                  const float* bias, int K, int N, float* out, int nsplit) {
    int klen = (((K + nsplit - 1) / nsplit) + 3) & ~3;
    int ns = (K + klen - 1) / klen;
    dim3 g(N / 16, ns);
    k_gemm_wmma<<<g, 32, 0, stream>>>(A, lda, W, N, K, klen, part);
    int tot = 16 * N;
    k_reduce_bias_relu<<<(tot + 255) / 256, 256, 0, stream>>>(part, ns, bias, N, out);
  };

  k_pack_obs<<<(16 * 8192 + 255) / 256, 256, 0, stream>>>(obs, OBS);
  k_pack_pad<<<(16 * 1024 + 255) / 256, 256, 0, stream>>>(LI, LIP, 12, 1024);
  k_pack_pad<<<(16 * 36 + 255) / 256, 256, 0, stream>>>(tf, TFP, 1, 36);

  gemm(OBS, 8192, fc_w1, fc_b1, 8192, 2048, H1, 16);
  gemm(H1, 2048, fc_w2, fc_b2, 2048, 2048, H2, 8);
  gemm(H2, 2048, fc_w3, fc_b3, 2048, 12288, H3, 4);
  gemm(LIP, 1024, gs_w1, gs_b1, 1024, 256, SP1, 8);
  gemm(SP1, 256, gs_w2, gs_b2, 256, 12288, SPU, 2);
  gemm(TFP, 36, gt_w1, gt_b1, 36, 256, TP1, 1);
  gemm(TP1, 256, gt_w2, gt_b2, 256, 12288, TPU, 2);

  k_build_gv<<<(3 * 12 * 1024 + 12 * 1024 + 255) / 256, 256, 0, stream>>>(
      SPU, TPU, Bm, GT, VCT);
  k_build_y<<<(3 * 1024 * 16 + 255) / 256, 256, 0, stream>>>(H3, Wf, Wb, Yb);
  k_adj_softmax_mm<<<48, 128, 0, stream>>>(GT, VCT, Yb, Pb);
  k_epilogue<<<(4 * 12288 + 255) / 256, 256, 0, stream>>>(Pb, bvec, (float*)d_out);
}